// TorchFurnaceSimulator_15187004359096
// MI455X (gfx1250) — compile-verified
//
#include <hip/hip_runtime.h>
#include <hip/hip_bf16.h>

// ---------------------------------------------------------------------------
// 2D diffusion (5-point Laplacian, Dirichlet-0 boundary), 400 steps.
//   u_{t+1} = mask * (u_t + 0.225*lap(u_t) + 1e-7*a)
//
// MI455X strategy: fields are 9 MB -> fully L2-resident (192 MB L2); the
// problem is bound by the 400 sequential syncs, not flops or HBM. Temporal
// blocking: each kernel advances 8 steps inside LDS (80x80 tile -> 64x64
// interior), so only 50 dispatches. Tile staging uses the gfx1250 async
// global->LDS engine (ASYNCcnt / s_wait_asynccnt); the forcing-term loads and
// mask setup are overlapped with the in-flight DMA before the wait.
// ---------------------------------------------------------------------------

#define GH     1501       // grid height/width (num_grid+1)
#define SPAR   0.225f     // diff*dt*num_grid^2
#define DTF    1e-7f      // delta_t
#define TSTEPS 8          // time steps fused per launch
#define BINT   64         // interior tile written per launch
#define BLOAD  80         // loaded tile = BINT + 2*TSTEPS
#define NTILE  24         // ceil(1501/64)
#define LSTR   82         // padded LDS row stride (1-ring pad each side)
#define LSZ    (LSTR*LSTR)
#define WPITCH 1536       // row pitch of workspace ping-pong buffers
#define NLAUNCH 50        // 400 / TSTEPS

#ifndef __has_builtin
#define __has_builtin(x) 0
#endif

#if defined(__HIP_DEVICE_COMPILE__) && \
    __has_builtin(__builtin_amdgcn_global_load_async_to_lds_b32) && \
    __has_builtin(__builtin_amdgcn_s_wait_asynccnt)
#define USE_ASYNC_LD 1
#else
#define USE_ASYNC_LD 0
#endif

#if defined(__HIP_DEVICE_COMPILE__) && \
    __has_builtin(__builtin_amdgcn_global_store_async_from_lds_b32)
#define USE_ASYNC_ST 1
#else
#define USE_ASYNC_ST 0
#endif

#if defined(__HIP_DEVICE_COMPILE__)
typedef __attribute__((address_space(1))) int GASI;  // global int
typedef __attribute__((address_space(3))) int LASI;  // LDS int
#endif

__global__ __launch_bounds__(256) void diffusion_tb8_kernel(
    const float* __restrict__ src, int sp,     // source field + row pitch
    float* __restrict__ dst, int dp,           // dest field + row pitch
    const float* __restrict__ a)               // forcing, pitch GH
{
    // two ping-pong 82x82 (padded 80x80) tiles: 53.8 KB of the 320 KB LDS
    __shared__ float lds[2 * LSZ];

    const int tid = threadIdx.x;
    const int ox  = blockIdx.x * BINT;     // interior origin (global col)
    const int oy  = blockIdx.y * BINT;     // interior origin (global row)
    const int gx0 = ox - TSTEPS;           // loaded-region origin
    const int gy0 = oy - TSTEPS;

    // ---- Phase 1a: issue async DMA of the 80x80 tile into lds[0..LSZ).
    // Out-of-domain cells get clamped (stale) values: provably never read by
    // any computation whose result is kept (those cells are zero-masked
    // before anyone reads them).
#pragma unroll
    for (int p = 0; p < 25; ++p) {
        int pos = p * 256 + tid;                 // 6400 = 80*80
        int ly = pos / BLOAD, lx = pos % BLOAD;
        int gi = gy0 + ly, gj = gx0 + lx;
        int cgi = min(max(gi, 0), GH - 1);
        int cgj = min(max(gj, 0), GH - 1);
        const float* gp = src + (size_t)cgi * sp + cgj;
        float* lp = &lds[(ly + 1) * LSTR + (lx + 1)];
#if USE_ASYNC_LD
        __builtin_amdgcn_global_load_async_to_lds_b32(
            (GASI*)gp, (LASI*)lp, 0, 0);
#else
        *lp = *gp;
#endif
    }

    // ---- Phase 1b (overlapped with the DMA): per-thread 5x5 ownership,
    // forcing registers, and 25-bit zero mask (boundary / out-of-domain).
    const int ty = tid >> 4, tx = tid & 15;
    const int r0 = ty * 5, c0 = tx * 5;        // loaded-region coords

    float fa[5][5];
    unsigned zmask = 0;
#pragma unroll
    for (int i = 0; i < 5; ++i) {
#pragma unroll
        for (int j = 0; j < 5; ++j) {
            int gi = gy0 + r0 + i, gj = gx0 + c0 + j;
            if (gi <= 0 || gi >= GH - 1 || gj <= 0 || gj >= GH - 1)
                zmask |= 1u << (i * 5 + j);
            int cgi = min(max(gi, 0), GH - 1);
            int cgj = min(max(gj, 0), GH - 1);
            fa[i][j] = DTF * a[(size_t)cgi * GH + cgj];
        }
    }

    // ---- DMA completion + workgroup visibility.
#if USE_ASYNC_LD
    __builtin_amdgcn_s_wait_asynccnt(0);
#endif
    __syncthreads();

    // ---- Phase 2: 8 Jacobi steps entirely in LDS (ping-pong halves).
    for (int s = 0; s < TSTEPS; ++s) {
        const int rOff = (s & 1) ? LSZ : 0;    // read half
        const int wOff = (s & 1) ? 0 : LSZ;    // write half

        float rA[7], rB[7], rC[7];
#pragma unroll
        for (int j = 0; j < 7; ++j) {          // loaded row r0-1 (padded r0)
            rA[j] = lds[rOff + r0 * LSTR + c0 + j];
            rB[j] = lds[rOff + (r0 + 1) * LSTR + c0 + j];
        }
#pragma unroll
        for (int i = 0; i < 5; ++i) {
#pragma unroll
            for (int j = 0; j < 7; ++j)        // loaded row r0+i+1
                rC[j] = lds[rOff + (r0 + i + 2) * LSTR + c0 + j];
#pragma unroll
            for (int j = 0; j < 5; ++j) {
                float c   = rB[j + 1];
                float lap = rA[j + 1] + rC[j + 1] + rB[j] + rB[j + 2] - 4.0f * c;
                float cand = fmaf(SPAR, lap, c) + fa[i][j];
                bool z = (zmask >> (i * 5 + j)) & 1;
                lds[wOff + (r0 + i + 1) * LSTR + (c0 + j + 1)] = z ? 0.0f : cand;
            }
#pragma unroll
            for (int j = 0; j < 7; ++j) { rA[j] = rB[j]; rB[j] = rC[j]; }
        }
        __syncthreads();
    }
    // After 8 steps (even), result sits in half 0.

    // ---- Phase 3: write back 64x64 interior (clip to domain) via async
    // LDS->global stores; S_ENDPGM implicitly drains ASYNCcnt.
#pragma unroll
    for (int p = 0; p < 16; ++p) {
        int pos = p * 256 + tid;               // 4096 = 64*64
        int iy = pos >> 6, ix = pos & 63;
        int gi = oy + iy, gj = ox + ix;
        if (gi < GH && gj < GH) {
            float* lp = &lds[(TSTEPS + iy + 1) * LSTR + (TSTEPS + ix + 1)];
            float* gp = dst + (size_t)gi * dp + gj;
#if USE_ASYNC_ST
            __builtin_amdgcn_global_store_async_from_lds_b32(
                (GASI*)gp, (LASI*)lp, 0, 0);
#else
            *gp = *lp;
#endif
        }
    }
}

extern "C" void kernel_launch(void* const* d_in, const int* in_sizes, int n_in,
                              void* d_out, int out_size, void* d_ws, size_t ws_size,
                              hipStream_t stream) {
    const float* u0 = (const float*)d_in[0];
    const float* a  = (const float*)d_in[1];
    float* out = (float*)d_out;

    // two pitched ping-pong buffers in workspace (9.24 MB each)
    float* b0 = (float*)d_ws;
    float* b1 = b0 + (size_t)WPITCH * 1504;

    dim3 grid(NTILE, NTILE), blk(256);
    for (int k = 0; k < NLAUNCH; ++k) {
        const float* src = (k == 0) ? u0 : ((k & 1) ? b0 : b1);
        int sp           = (k == 0) ? GH : WPITCH;
        float* dst       = (k == NLAUNCH - 1) ? out : ((k & 1) ? b1 : b0);
        int dp           = (k == NLAUNCH - 1) ? GH : WPITCH;
        hipLaunchKernelGGL(diffusion_tb8_kernel, grid, blk, 0, stream,
                           src, sp, dst, dp, a);
    }
}